// ClusterNorm1dv2_37151467110658
// MI455X (gfx1250) — compile-verified
//
#include <hip/hip_runtime.h>
#include <hip/hip_bf16.h>

// ClusterNorm1dv2 forward for MI455X (gfx1250, wave32, WMMA).
// Phase 1: raw moments S=sum x x^T, s=sum x per cluster via V_WMMA_F32_16X16X4_F32 SYRK.
// Phase 2: covariance + Cholesky + explicit L^-1 and v = L^-1 mu (per-cluster, 1 wave).
// Phase 3: Z = L^-1 x - v as WMMA GEMM, LDS-staged coalesced stores.

#define BATCH   16384
#define DIM     32
#define NK      128
#define KPB     16                    // clusters per block
#define KG      (NK / KPB)            // 8
#define TILE_B  32
#define DSTR    17                    // padded d-stride (dwords) -> conflict-free operand reads
#define BSTR    (DIM * DSTR + 1)      // 545 padded batch-stride (dwords)
#define TILE_DW (TILE_B * BSTR)       // 17440 dwords = 69.76 KB per buffer
#define P3_TILES 4
#define P3_CHUNKS (BATCH / (TILE_B * P3_TILES))   // 128

typedef __attribute__((ext_vector_type(2))) float v2f;
typedef __attribute__((ext_vector_type(8))) float v8f;

#if defined(__gfx1250__) && __has_builtin(__builtin_amdgcn_global_load_async_to_lds_b32)
#define HAVE_ASYNC 1
#pragma message("CDNA5: using global_load_async_to_lds_b32 async path")
#else
#define HAVE_ASYNC 0
#pragma message("CDNA5: async-to-LDS builtin unavailable in this pass; sync fallback")
#endif

#if HAVE_ASYNC
#if __has_builtin(__builtin_amdgcn_s_wait_asynccnt)
#define WAIT_ASYNC(n) __builtin_amdgcn_s_wait_asynccnt(n)
#else
#define WAIT_ASYNC(n) asm volatile("s_wait_asynccnt %0" ::"n"(n) : "memory")
#endif
#else
#define WAIT_ASYNC(n) ((void)0)
#endif

#if HAVE_ASYNC
typedef __attribute__((address_space(1))) int as1_int;
typedef __attribute__((address_space(3))) int as3_int;
#endif

__device__ __forceinline__ void cp_async_f32(const float* g, float* l) {
#if HAVE_ASYNC
  __builtin_amdgcn_global_load_async_to_lds_b32(
      (as1_int*)g, (as3_int*)l, 0, 0);
#else
  *l = *g;
#endif
}

__device__ __forceinline__ v8f wmma4(v2f a, v2f b, v8f c) {
#if defined(__HIP_DEVICE_COMPILE__)
  // D(16x16 f32) = A(16x4 f32) * B(4x16 f32) + C
  return __builtin_amdgcn_wmma_f32_16x16x4_f32(false, a, false, b, (short)0, c,
                                               false, false);
#else
  (void)a; (void)b;
  return c;
#endif
}

// ---------------------------------------------------------------------------
// Phase 1: per-cluster raw second moment (SYRK) + raw sum, split over batch.
// grid = (KG, nsplit), block = 512 (16 waves, wave w -> cluster k0+w)
// ---------------------------------------------------------------------------
__global__ __launch_bounds__(512, 1)
void p1_moments(const float* __restrict__ x, float* __restrict__ Spart,
                float* __restrict__ spart, int nsplit) {
  __shared__ float xt[2][TILE_DW];
  const int tid   = threadIdx.x;
  const int lane  = tid & 31;
  const int wave  = tid >> 5;
  const int kg    = blockIdx.x;
  const int split = blockIdx.y;
  const int k0    = kg * KPB;
  const int bpb   = BATCH / nsplit;
  const int ntiles = bpb / TILE_B;
  const int bbase  = split * bpb;

  // loader mapping: thread -> (d, k) element, loops over the 32 batch rows
  const int ld_d = tid >> 4;     // 0..31
  const int ld_k = tid & 15;     // 0..15
  const float* gbase =
      x + (((size_t)bbase * DIM + ld_d) * NK + (size_t)k0 + ld_k);

  auto issue_tile = [&](int buf, int t) {
    const float* gp = gbase + (size_t)t * TILE_B * (DIM * NK);
    float* lp = &xt[buf][ld_d * DSTR + ld_k];
#pragma unroll
    for (int b = 0; b < TILE_B; ++b)
      cp_async_f32(gp + (size_t)b * (DIM * NK), lp + b * BSTR);
  };

  const int half = lane >> 4;    // 0/1 -> contraction batches {0,1} vs {2,3}
  const int m    = lane & 15;    // matrix row / col within 16-tile
  const int kl   = wave;         // cluster column in LDS tile

  v8f acc00 = {};  // S[0:16, 0:16]
  v8f acc01 = {};  // S[0:16,16:32]  (tile10 is its transpose)
  v8f acc11 = {};  // S[16:32,16:32]
  float vs0 = 0.f, vs1 = 0.f;

  issue_tile(0, 0);
  for (int t = 0; t < ntiles; ++t) {
    const int buf = t & 1;
    if (t + 1 < ntiles) { issue_tile(buf ^ 1, t + 1); WAIT_ASYNC(TILE_B); }
    else                { WAIT_ASYNC(0); }
    __syncthreads();
    const float* Lb = xt[buf];
    const int base = m * DSTR + kl + 2 * half * BSTR;
#pragma unroll
    for (int s = 0; s < 8; ++s) {
      const int i0 = s * 4 * BSTR + base;
      v2f a0, a1;
      a0.x = Lb[i0];              a0.y = Lb[i0 + BSTR];
      a1.x = Lb[i0 + 16 * DSTR];  a1.y = Lb[i0 + BSTR + 16 * DSTR];
      acc00 = wmma4(a0, a0, acc00);
      acc01 = wmma4(a0, a1, acc01);
      acc11 = wmma4(a1, a1, acc11);
      vs0 += a0.x + a0.y;   // raw sum fold-in (rows 0..15)
      vs1 += a1.x + a1.y;   // raw sum fold-in (rows 16..31)
    }
    __syncthreads();
  }

  float* Sp = Spart + ((size_t)split * NK + (k0 + kl)) * 1024;
#pragma unroll
  for (int r = 0; r < 8; ++r) {
    const int mm = r + 8 * half;               // C/D layout: M = r (+8 for hi lanes)
    Sp[mm * 32 + m]              = acc00[r];
    Sp[mm * 32 + 16 + m]         = acc01[r];
    Sp[(16 + m) * 32 + mm]       = acc01[r];   // mirrored lower tile
    Sp[(16 + mm) * 32 + 16 + m]  = acc11[r];
  }
  float* sp = spart + ((size_t)split * NK + (k0 + kl)) * 64;
  sp[lane]      = vs0;   // row (lane&15), two lane-halves to be summed later
  sp[32 + lane] = vs1;   // row 16+(lane&15)
}

// ---------------------------------------------------------------------------
// Phase 2: per-cluster covariance, Cholesky, explicit inverse, v = Linv*mu.
// grid = NK, block = 32 (one wave, lane = matrix row / solve column)
// ---------------------------------------------------------------------------
__global__ __launch_bounds__(32, 1)
void p2_prepare(const float* __restrict__ Spart, const float* __restrict__ spart,
                const float* __restrict__ mu0g, const float* __restrict__ L0g,
                const float* __restrict__ n0g, float* __restrict__ Linvg,
                float* __restrict__ vvecg, int nsplit) {
  __shared__ float Ms[32][33];
  __shared__ float Ls[32][33];
  __shared__ float Ys[32][33];
  __shared__ float xmu_s[32], dmu_s[32], nm_s[32];
  const int k = blockIdx.x;
  const int r = threadIdx.x;

  // reduce raw sum -> x_mu
  float sv = 0.f;
  for (int sp = 0; sp < nsplit; ++sp) {
    const float* s64 = spart + ((size_t)sp * NK + k) * 64;
    sv += (r < 16) ? (s64[r] + s64[r + 16]) : (s64[16 + r] + s64[32 + r]);
  }
  const float Bf  = (float)BATCH;
  const float xmu = sv / Bf;
  const float n0  = n0g[0];
  const float den = n0 + Bf;
  const float mu0 = mu0g[r * NK + k];
  xmu_s[r] = xmu;
  dmu_s[r] = xmu - mu0;
  nm_s[r]  = (n0 * mu0 + Bf * xmu) / den;
  for (int c = 0; c < 32; ++c)
    Ls[r][c] = L0g[(size_t)k * 1024 + r * 32 + c];   // stage L_0
  __syncthreads();

  // M = (n0/den) L0 L0^T + (S - B mu mu^T)/den + (n0 B/den^2) dmu dmu^T + I
  const float cA = n0 / den, cC = 1.f / den, cD = n0 * Bf / (den * den);
  const float dmur = xmu - mu0;
  for (int c = 0; c < 32; ++c) {
    float s = 0.f;
    for (int sp = 0; sp < nsplit; ++sp)
      s += Spart[((size_t)sp * NK + k) * 1024 + r * 32 + c];
    float g = 0.f;
#pragma unroll 8
    for (int t = 0; t < 32; ++t) g += Ls[r][t] * Ls[c][t];
    const float Cc = s - Bf * xmu * xmu_s[c];
    Ms[r][c] = cA * g + cC * Cc + cD * dmur * dmu_s[c] + ((r == c) ? 1.f : 0.f);
  }
  __syncthreads();

  // Cholesky (right-looking), lanes = rows
  for (int j = 0; j < 32; ++j) {
    const float diag = sqrtf(Ms[j][j]);
    const float lrj  = (r >= j) ? (Ms[r][j] / diag) : 0.f;
    Ls[r][j] = lrj;
    __syncthreads();
    for (int c = j + 1; c <= r; ++c) Ms[r][c] -= lrj * Ls[c][j];
    __syncthreads();
  }

  // Linv: lane r forward-solves column c=r of L Y = I (no cross-lane deps)
  {
    const int c = r;
    for (int rr = 0; rr < 32; ++rr) {
      float a = (rr == c) ? 1.f : 0.f;
      for (int t = c; t < rr; ++t) a -= Ls[rr][t] * Ys[t][c];
      Ys[rr][c] = (rr >= c) ? (a / Ls[rr][rr]) : 0.f;
    }
  }
  __syncthreads();

  // v = Linv * new_mu, then store Linv row-major
  float vv = 0.f;
  for (int t = 0; t <= r; ++t) vv += Ys[r][t] * nm_s[t];
  vvecg[k * 32 + r] = vv;
  for (int c = 0; c < 32; ++c)
    Linvg[(size_t)k * 1024 + r * 32 + c] = Ys[r][c];
}

// ---------------------------------------------------------------------------
// Phase 3: Z = Linv * x - v  per cluster; WMMA GEMM over the batch stream.
// grid = (KG, P3_CHUNKS), block = 512
// ---------------------------------------------------------------------------
__global__ __launch_bounds__(512, 1)
void p3_whiten(const float* __restrict__ x, const float* __restrict__ Linvg,
               const float* __restrict__ vvecg, float* __restrict__ out) {
  __shared__ float xt[2][TILE_DW];
  const int tid   = threadIdx.x;
  const int lane  = tid & 31;
  const int wave  = tid >> 5;
  const int kg    = blockIdx.x;
  const int chunk = blockIdx.y;
  const int k0    = kg * KPB;
  const int kl    = wave;
  const int k     = k0 + kl;
  const int half  = lane >> 4;
  const int m     = lane & 15;
  const int bbase = chunk * (TILE_B * P3_TILES);

  const int ld_d = tid >> 4;
  const int ld_k = tid & 15;
  const float* gbase =
      x + (((size_t)bbase * DIM + ld_d) * NK + (size_t)k0 + ld_k);
  auto issue_tile = [&](int buf, int t) {
    const float* gp = gbase + (size_t)t * TILE_B * (DIM * NK);
    float* lp = &xt[buf][ld_d * DSTR + ld_k];
#pragma unroll
    for (int b = 0; b < TILE_B; ++b)
      cp_async_f32(gp + (size_t)b * (DIM * NK), lp + b * BSTR);
  };

  // Preload Linv as WMMA A-operand chunks; top row-tile needs only cols 0..15
  v2f la0[4], la1[8];
  {
    const float* Lk = Linvg + (size_t)k * 1024 + 2 * half;  // kk offset per half
#pragma unroll
    for (int c2 = 0; c2 < 4; ++c2) {
      la0[c2].x = Lk[m * 32 + 4 * c2];
      la0[c2].y = Lk[m * 32 + 4 * c2 + 1];
    }
#pragma unroll
    for (int c2 = 0; c2 < 8; ++c2) {
      la1[c2].x = Lk[(16 + m) * 32 + 4 * c2];
      la1[c2].y = Lk[(16 + m) * 32 + 4 * c2 + 1];
    }
  }
  float vt0[8], vt1[8];
#pragma unroll
  for (int r2 = 0; r2 < 8; ++r2) {
    vt0[r2] = vvecg[k * 32 + 8 * half + r2];
    vt1[r2] = vvecg[k * 32 + 16 + 8 * half + r2];
  }

  issue_tile(0, 0);
  for (int t = 0; t < P3_TILES; ++t) {
    const int buf = t & 1;
    if (t + 1 < P3_TILES) { issue_tile(buf ^ 1, t + 1); WAIT_ASYNC(TILE_B); }
    else                  { WAIT_ASYNC(0); }
    __syncthreads();
    float* Lb = xt[buf];

    v8f A0[2], A1[2];
#pragma unroll
    for (int s = 0; s < 2; ++s) {
      v8f a0 = {}, a1 = {};
      const int nb = (s * 16 + m) * BSTR + 2 * half * DSTR + kl;
#pragma unroll
      for (int c2 = 0; c2 < 8; ++c2) {
        v2f bop;
        bop.x = Lb[nb + (4 * c2) * DSTR];
        bop.y = Lb[nb + (4 * c2 + 1) * DSTR];
        if (c2 < 4) a0 = wmma4(la0[c2], bop, a0);
        a1 = wmma4(la1[c2], bop, a1);
      }
      A0[s] = a0; A1[s] = a1;
    }
    __syncthreads();           // all waves done reading xt[buf]

    // reuse xt[buf] as output staging: Z = acc - v
#pragma unroll
    for (int s = 0; s < 2; ++s) {
#pragma unroll
      for (int r2 = 0; r2 < 8; ++r2) {
        const int d0 = r2 + 8 * half;
        Lb[(s * 16 + m) * BSTR + d0 * DSTR + kl]        = A0[s][r2] - vt0[r2];
        Lb[(s * 16 + m) * BSTR + (16 + d0) * DSTR + kl] = A1[s][r2] - vt1[r2];
      }
    }
    __syncthreads();

    // cooperative coalesced store back to [B, D, K]
    float* ob = out + (((size_t)(bbase + t * TILE_B) * DIM + ld_d) * NK +
                       (size_t)k0 + ld_k);
    const float* lp = &Lb[ld_d * DSTR + ld_k];
#pragma unroll
    for (int b = 0; b < TILE_B; ++b)
      ob[(size_t)b * (DIM * NK)] = lp[b * BSTR];
    __syncthreads();
  }
}

// ---------------------------------------------------------------------------
extern "C" void kernel_launch(void* const* d_in, const int* in_sizes, int n_in,
                              void* d_out, int out_size, void* d_ws, size_t ws_size,
                              hipStream_t stream) {
  (void)in_sizes; (void)n_in; (void)out_size;
  const float* x   = (const float*)d_in[0];
  const float* mu0 = (const float*)d_in[1];
  const float* L0  = (const float*)d_in[2];
  const float* n0  = (const float*)d_in[3];
  float* out = (float*)d_out;
  float* ws  = (float*)d_ws;

  // pick largest power-of-two split whose partials fit in the workspace
  int nsplit = 32;
  while (nsplit > 1) {
    size_t need = ((size_t)nsplit * NK * (1024 + 64) + (size_t)NK * 1024 +
                   (size_t)NK * 32) * sizeof(float);
    if (need <= ws_size) break;
    nsplit >>= 1;
  }

  float* Spart = ws;
  float* spart = Spart + (size_t)nsplit * NK * 1024;
  float* Linv  = spart + (size_t)nsplit * NK * 64;
  float* vvec  = Linv  + (size_t)NK * 1024;

  p1_moments<<<dim3(KG, nsplit), 512, 0, stream>>>(x, Spart, spart, nsplit);
  p2_prepare<<<dim3(NK), 32, 0, stream>>>(Spart, spart, mu0, L0, n0, Linv, vvec,
                                          nsplit);
  p3_whiten<<<dim3(KG, P3_CHUNKS), 512, 0, stream>>>(x, Linv, vvec, out);
}